// TemporalMemoryTokens_85736137162898
// MI455X (gfx1250) — compile-verified
//
#include <hip/hip_runtime.h>

// ---------------------------------------------------------------------------
// TemporalMemoryTokens, fused single-kernel CDNA5 implementation.
//
// Softmax over a length-1 axis is identically 1 => q/k/attention are dead.
// Per step:
//   v = Wv f + bv ; u = Wo v + bo ; c = Wg2 u + bg      (frame-only)
//   per mem row:  g = sigmoid(Wg1 m + c); m = LN(g*u + (1-g)*m)
// One workgroup = 16 batches (128 mem rows), whole T-loop in-kernel, state in
// LDS, fp32 WMMA (v_wmma_f32_16x16x4_f32) for every matvec.  Frame tiles are
// double-buffered via GLOBAL_LOAD_ASYNC_TO_LDS_B128 (ASYNCcnt), so the only
// per-step global traffic (16 KB of frames) is hidden behind the WMMA work.
// ---------------------------------------------------------------------------

typedef float v2f __attribute__((ext_vector_type(2)));
typedef float v8f __attribute__((ext_vector_type(8)));

namespace {
constexpr int kD          = 256;
constexpr int kT          = 64;
constexpr int kTok        = 8;
constexpr int kBatchPerWG = 16;
constexpr int kMemRows    = kBatchPerWG * kTok;   // 128
constexpr int kStride     = 260;                  // padded LDS row stride (floats)
}

__device__ __forceinline__ v8f wmma_f32(v2f a, v2f b, v8f c) {
  // D = A(16x4) * B(4x16) + C(16x16), fp32
  return __builtin_amdgcn_wmma_f32_16x16x4_f32(
      /*neg_a=*/false, a, /*neg_b=*/false, b,
      /*c_mod=*/(short)0, c, /*reuse_a=*/false, /*reuse_b=*/false);
}

// Asynchronously stage one 16x256 frame tile (step t) into an LDS buffer.
// 4 x b128 per thread; tracked by ASYNCcnt, completed via s_wait_asynccnt.
__device__ __forceinline__ void stage_frames_async(
    const float* __restrict__ gsrc,   // frames + per-WG base
    float* __restrict__ dstbuf,       // LDS tile [16][kStride]
    int tid, int t)
{
#pragma unroll
  for (int i = 0; i < 4; ++i) {
    const int idx4 = tid + i * 256;   // [16 rows][64 float4]
    const int row = idx4 >> 6, c4 = idx4 & 63;
    const float* g = gsrc + ((size_t)row * kT + t) * kD + c4 * 4;
    const unsigned lds = (unsigned)(size_t)(&dstbuf[row * kStride + c4 * 4]);
    asm volatile("global_load_async_to_lds_b128 %0, %1, off"
                 :: "v"(lds), "v"((unsigned long long)(size_t)g)
                 : "memory");
  }
}

__device__ __forceinline__ void wait_async_all() {
  asm volatile("s_wait_asynccnt 0x0" ::: "memory");
}

// Y[16][D] = A[16][D] @ W^T + bias, for this wave's two 16-wide N tiles.
// A, Y are LDS tiles with row stride kStride; W is global row-major with
// row stride wstride (W[n][k] = W + n*wstride + k).
__device__ __forceinline__ void gemm16_2tiles(
    const float* __restrict__ As, const float* __restrict__ W, int wstride,
    const float* __restrict__ bias, float* __restrict__ Ys,
    int wave, int lm, int lh)
{
  const int nt0 = 2 * wave, nt1 = nt0 + 1;
  const v8f zero = {0.f,0.f,0.f,0.f,0.f,0.f,0.f,0.f};
  v8f acc0 = zero, acc1 = zero;
  const float* w0   = W  + (size_t)(nt0 * 16 + lm) * wstride + 2 * lh;
  const float* w1   = W  + (size_t)(nt1 * 16 + lm) * wstride + 2 * lh;
  const float* arow = As + lm * kStride + 2 * lh;
#pragma unroll 8
  for (int kk = 0; kk < 64; ++kk) {
    v2f a  = *(const v2f*)(arow + 4 * kk);     // ds_load_b64, bank-conflict-free
    v2f b0 = *(const v2f*)(w0 + 4 * kk);       // L2-resident weights
    v2f b1 = *(const v2f*)(w1 + 4 * kk);
    acc0 = wmma_f32(a, b0, acc0);
    acc1 = wmma_f32(a, b1, acc1);
  }
  const float bval0 = bias[nt0 * 16 + lm];
  const float bval1 = bias[nt1 * 16 + lm];
#pragma unroll
  for (int r = 0; r < 8; ++r) {
    const int row = r + 8 * lh;                // C layout: vgpr r, lane half
    Ys[row * kStride + nt0 * 16 + lm] = acc0[r] + bval0;
    Ys[row * kStride + nt1 * 16 + lm] = acc1[r] + bval1;
  }
}

__global__ __launch_bounds__(256)
void tmt_fused_kernel(const float* __restrict__ frames,
                      const float* __restrict__ memory_init,
                      const float* __restrict__ Wv, const float* __restrict__ bv,
                      const float* __restrict__ Wo, const float* __restrict__ bo,
                      const float* __restrict__ Wg, const float* __restrict__ bg,
                      const float* __restrict__ ln_w, const float* __restrict__ ln_b,
                      float* __restrict__ out)
{
  __shared__ float s_fr [2][16 * kStride];     //  double-buffered frame tiles
  __shared__ float s_v  [16 * kStride];        //  v = Wv f + bv
  __shared__ float s_u  [16 * kStride];        //  u = Wo v + bo
  __shared__ float s_c  [16 * kStride];        //  c = Wg2 u + bg
  __shared__ float s_mem[kMemRows * kStride];  //  memory state [128][256]

  const int tid  = threadIdx.x;
  const int wave = tid >> 5;
  const int lane = tid & 31;
  const int lm   = lane & 15;   // M/N index within a 16-tile
  const int lh   = lane >> 4;   // lane half selects K pair / row+8
  const int wg   = blockIdx.x;

  const float* gframes = frames + (size_t)wg * kBatchPerWG * kT * kD;

  // ---- init memory tokens (same init replicated across batches) ----------
  for (int idx = tid; idx < kMemRows * kD; idx += 256) {
    const int row = idx >> 8, col = idx & 255;
    s_mem[row * kStride + col] = memory_init[(row & 7) * kD + col];
  }
  // per-lane LayerNorm params for columns lane*8 .. lane*8+7
  float lnw[8], lnb[8];
#pragma unroll
  for (int j = 0; j < 8; ++j) {
    lnw[j] = ln_w[lane * 8 + j];
    lnb[j] = ln_b[lane * 8 + j];
  }
  // kick off async staging of the t=0 frame tile
  stage_frames_async(gframes, s_fr[0], tid, 0);

  const v8f zero = {0.f,0.f,0.f,0.f,0.f,0.f,0.f,0.f};

  for (int t = 0; t < kT; ++t) {
    // ---- 0. frame tile for step t is ready after wait + barrier ---------
    wait_async_all();       // this wave's async LDS writes complete
    __syncthreads();        // everyone's writes visible (also covers LN of t-1)

    // prefetch step t+1 tile into the other buffer, hidden behind compute
    if (t + 1 < kT)
      stage_frames_async(gframes, s_fr[(t + 1) & 1], tid, t + 1);

    const float* fr = s_fr[t & 1];

    // ---- 1..3. chained frame matvecs via fp32 WMMA -----------------------
    gemm16_2tiles(fr, Wv, kD, bv, s_v, wave, lm, lh);          // v
    __syncthreads();
    gemm16_2tiles(s_v, Wo, kD, bo, s_u, wave, lm, lh);         // u
    __syncthreads();
    gemm16_2tiles(s_u, Wg + kD, 2 * kD, bg, s_c, wave, lm, lh);// c (Wg2 slice)
    __syncthreads();

    // ---- 4. gate GEMM: y = mem_tile @ Wg1^T, then gated update ----------
    {
      v8f acc[16];
#pragma unroll
      for (int n = 0; n < 16; ++n) acc[n] = zero;
      const float* arow = &s_mem[(wave * 16 + lm) * kStride + 2 * lh];
#pragma unroll 4
      for (int kk = 0; kk < 64; ++kk) {
        const v2f a = *(const v2f*)(arow + 4 * kk);
#pragma unroll
        for (int n = 0; n < 16; ++n) {
          const v2f b = *(const v2f*)(Wg + (size_t)(n * 16 + lm) * (2 * kD) +
                                      4 * kk + 2 * lh);        // Wg1 slice
          acc[n] = wmma_f32(a, b, acc[n]);
        }
      }
#pragma unroll
      for (int n = 0; n < 16; ++n) {
        const int col = n * 16 + lm;
#pragma unroll
        for (int r = 0; r < 8; ++r) {
          const int row = wave * 16 + r + 8 * lh;  // this wave's rows only
          const int bat = row >> 3;                // local batch 0..15
          const float y  = acc[n][r] + s_c[bat * kStride + col];
          const float g  = 1.0f / (1.0f + __expf(-y));
          const float uu = s_u[bat * kStride + col];
          const float m0 = s_mem[row * kStride + col];
          s_mem[row * kStride + col] = g * uu + (1.0f - g) * m0;
        }
      }
    }

    // ---- 5. LayerNorm in place on this wave's 16 rows -------------------
    // (wave-private rows: no barrier needed before the next top-of-step sync)
#pragma unroll 2
    for (int r = 0; r < 16; ++r) {
      float* rowp = &s_mem[(wave * 16 + r) * kStride + lane * 8];
      float x[8];
      *(float4*)&x[0] = *(const float4*)rowp;        // ds_load_b128
      *(float4*)&x[4] = *(const float4*)(rowp + 4);
      float s = 0.f, q = 0.f;
#pragma unroll
      for (int j = 0; j < 8; ++j) { s += x[j]; q += x[j] * x[j]; }
#pragma unroll
      for (int off = 16; off > 0; off >>= 1) {       // wave32 tree reduce
        s += __shfl_xor(s, off, 32);
        q += __shfl_xor(q, off, 32);
      }
      const float mu  = s * (1.0f / 256.0f);
      const float var = q * (1.0f / 256.0f) - mu * mu;
      const float rs  = rsqrtf(var + 1e-5f);
#pragma unroll
      for (int j = 0; j < 8; ++j) x[j] = (x[j] - mu) * rs * lnw[j] + lnb[j];
      *(float4*)rowp       = *(const float4*)&x[0];
      *(float4*)(rowp + 4) = *(const float4*)&x[4];
    }
  }

  __syncthreads();
  // ---- final: write memory tokens [B, N, D] -----------------------------
  for (int idx = tid; idx < kMemRows * kD; idx += 256) {
    const int row = idx >> 8, col = idx & 255;
    out[(size_t)wg * kMemRows * kD + idx] = s_mem[row * kStride + col];
  }
}

extern "C" void kernel_launch(void* const* d_in, const int* in_sizes, int n_in,
                              void* d_out, int out_size, void* d_ws, size_t ws_size,
                              hipStream_t stream) {
  // setup_inputs order:
  // 0 frames, 1 memory_init, 2 Wq, 3 bq, 4 Wk, 5 bk, 6 Wv, 7 bv,
  // 8 Wo, 9 bo, 10 Wg, 11 bg, 12 ln_w, 13 ln_b   (Wq/bq/Wk/bk are dead)
  const float* frames      = (const float*)d_in[0];
  const float* memory_init = (const float*)d_in[1];
  const float* Wv  = (const float*)d_in[6];
  const float* bv  = (const float*)d_in[7];
  const float* Wo  = (const float*)d_in[8];
  const float* bo  = (const float*)d_in[9];
  const float* Wg  = (const float*)d_in[10];
  const float* bg  = (const float*)d_in[11];
  const float* lnw = (const float*)d_in[12];
  const float* lnb = (const float*)d_in[13];
  float* out = (float*)d_out;

  const int numB = in_sizes[0] / (kT * kD);          // 4096
  const int grid = numB / kBatchPerWG;               // 256 workgroups

  tmt_fused_kernel<<<dim3(grid), dim3(256), 0, stream>>>(
      frames, memory_init, Wv, bv, Wo, bo, Wg, bg, lnw, lnb, out);
}